// RandomWalkPositionalEncoding_3959959847625
// MI455X (gfx1250) — compile-verified
//
#include <hip/hip_runtime.h>

#define NODES 100000
#define WALK  16
#define EMBED 64

typedef __attribute__((ext_vector_type(2))) float v2f;
typedef __attribute__((ext_vector_type(8))) float v8f;

// ---- degree / normalization ------------------------------------------------

__global__ void k_init_deg(float* __restrict__ deg, int n) {
  int i = blockIdx.x * blockDim.x + threadIdx.x;
  if (i < n) deg[i] = 1.0f;                    // self-loop counts once
}

__global__ void k_count_deg(const int* __restrict__ col, float* __restrict__ deg, int e) {
  int i = blockIdx.x * blockDim.x + threadIdx.x;
  if (i < e) atomicAdd(&deg[col[i]], 1.0f);
}

__global__ void k_finish_deg(float* __restrict__ deg, float* __restrict__ prob,
                             int n, float p0) {
  int i = blockIdx.x * blockDim.x + threadIdx.x;
  if (i < n) {
    deg[i]  = rsqrtf(deg[i]);                  // deg >= 1 always (self loops)
    prob[i] = p0;
  }
}

__global__ void k_norm(const int* __restrict__ row, const int* __restrict__ col,
                       const float* __restrict__ dinv, float* __restrict__ norm, int e) {
  int i = blockIdx.x * blockDim.x + threadIdx.x;
  if (i < e) norm[i] = dinv[row[i]] * dinv[col[i]];
}

// ---- random-walk iteration -------------------------------------------------

__global__ void k_save_zero(const float* __restrict__ prob, float* __restrict__ rw,
                            float* __restrict__ acc, int n, int t) {
  int i = blockIdx.x * blockDim.x + threadIdx.x;
  if (i < n) {
    rw[i * WALK + t] = prob[i];                // record prob BEFORE the step
    acc[i] = 0.0f;
  }
}

__global__ void k_spmv(const int* __restrict__ row, const int* __restrict__ col,
                       const float* __restrict__ norm, const float* __restrict__ prob,
                       float* __restrict__ acc, int e) {
  int i = blockIdx.x * blockDim.x + threadIdx.x;
  if (i < e) atomicAdd(&acc[col[i]], prob[row[i]] * norm[i]);
}

__global__ void k_combine(const float* __restrict__ acc, const float* __restrict__ dinv,
                          float* __restrict__ prob, int n) {
  int i = blockIdx.x * blockDim.x + threadIdx.x;
  if (i < n) {
    float p = prob[i];
    float d = dinv[i];
    float segsum = acc[i] + p * d * d;         // fold self-loop contribution
    prob[i] = 0.9f * segsum + 0.1f * p;
  }
}

// ---- final linear via V_WMMA_F32_16X16X4_F32 ------------------------------
// out[n][d] = sum_t rw[n][t] * W[d][t] + b[d]
// One wave32 computes one 16(nodes) x 16(embed) tile; K=16 via 4 chained WMMAs.

__global__ void k_linear_wmma(const float* __restrict__ rw, const float* __restrict__ W,
                              const float* __restrict__ bias, float* __restrict__ out) {
  int wave = (blockIdx.x * blockDim.x + threadIdx.x) >> 5;
  int lane = threadIdx.x & 31;
  int node_tile = wave >> 2;                   // EMBED/16 == 4 tiles per node group
  int emb_tile  = wave & 3;
  if (node_tile >= NODES / 16) return;         // whole-wave uniform exit
  int nb   = node_tile * 16;
  int db   = emb_tile * 16;
  int half = lane >> 4;                        // 0: K pair {0,1}, 1: K pair {2,3}
  int l    = lane & 15;

  v8f c = {};
#pragma unroll
  for (int s = 0; s < 4; ++s) {
    int kb = 4 * s + 2 * half;
    v2f a, bm;
    // A 16x4 f32: lane half selects K pair, VGPR index selects K within pair
    a.x = rw[(nb + l) * WALK + kb + 0];
    a.y = rw[(nb + l) * WALK + kb + 1];
    // B 4x16 f32: B[k][d] = W[d][k]; row k striped across lanes
    bm.x = W[(db + l) * WALK + kb + 0];
    bm.y = W[(db + l) * WALK + kb + 1];
    // 8 args: (neg_a, A, neg_b, B, c_mod, C, reuse_a, reuse_b)
    c = __builtin_amdgcn_wmma_f32_16x16x4_f32(false, a, false, bm,
                                              (short)0, c, false, false);
  }

  float bv = bias[db + l];
#pragma unroll
  for (int j = 0; j < 8; ++j) {
    int m = j + 8 * half;                      // C/D layout: M = j + 8*half, N = l
    out[(nb + m) * EMBED + db + l] = c[j] + bv;
  }
}

// ---- launcher --------------------------------------------------------------

extern "C" void kernel_launch(void* const* d_in, const int* in_sizes, int n_in,
                              void* d_out, int out_size, void* d_ws, size_t ws_size,
                              hipStream_t stream) {
  const int*   edge = (const int*)d_in[0];     // (2, E) flattened
  const int    E    = in_sizes[0] / 2;
  const int*   row  = edge;
  const int*   col  = edge + E;
  const float* W    = (const float*)d_in[2];   // (64, 16)
  const float* bias = (const float*)d_in[3];   // (64,)
  float*       out  = (float*)d_out;           // (N, 64)

  float* dinv = (float*)d_ws;                  // N
  float* prob = dinv + NODES;                  // N
  float* acc  = prob + NODES;                  // N
  float* rw   = acc + NODES;                   // N * 16
  float* norm = rw + (size_t)NODES * WALK;     // E

  const int B  = 256;
  const int gN = (NODES + B - 1) / B;
  const int gE = (E + B - 1) / B;

  k_init_deg  <<<gN, B, 0, stream>>>(dinv, NODES);
  k_count_deg <<<gE, B, 0, stream>>>(col, dinv, E);
  k_finish_deg<<<gN, B, 0, stream>>>(dinv, prob, NODES, 1.0f / (float)NODES);
  k_norm      <<<gE, B, 0, stream>>>(row, col, dinv, norm, E);

  for (int t = 0; t < WALK; ++t) {
    k_save_zero<<<gN, B, 0, stream>>>(prob, rw, acc, NODES, t);
    k_spmv     <<<gE, B, 0, stream>>>(row, col, norm, prob, acc, E);
    k_combine  <<<gN, B, 0, stream>>>(acc, dinv, prob, NODES);
  }

  const int tiles = (NODES / 16) * (EMBED / 16);   // 25000 wave-tiles
  const int wpb   = B / 32;                        // 8 waves per block
  k_linear_wmma<<<(tiles + wpb - 1) / wpb, B, 0, stream>>>(rw, W, bias, out);
}